// MPNN_73289321939187
// MI455X (gfx1250) — compile-verified
//
#include <hip/hip_runtime.h>
#include <hip/hip_bf16.h>
#include <math.h>

typedef __attribute__((ext_vector_type(2))) float v2f;
typedef __attribute__((ext_vector_type(8))) float v8f;

#define NN 50000
#define HD 128
#define KD 256
#define LDSS 260   // padded LDS row stride (floats) to break 64-bank conflicts

// ---------------- workspace zero ----------------
__global__ __launch_bounds__(256) void mpnn_zero(float4* __restrict__ p, int n4) {
    int i = blockIdx.x * blockDim.x + threadIdx.x;
    int stride = gridDim.x * blockDim.x;
    float4 z = {0.f, 0.f, 0.f, 0.f};
    for (; i < n4; i += stride) p[i] = z;
}

// ---------------- edge scatter: S[dst] += h[src], cnt[dst] += 1 ----------------
__global__ __launch_bounds__(256) void mpnn_scatter(const float* __restrict__ H,
                                                    const int* __restrict__ src,
                                                    const int* __restrict__ dst,
                                                    float* __restrict__ S,
                                                    float* __restrict__ cnt,
                                                    int E) {
    int gwave  = (blockIdx.x * 256 + threadIdx.x) >> 5;
    int lane   = threadIdx.x & 31;
    int nwaves = (gridDim.x * 256) >> 5;
    for (int e = gwave; e < E; e += nwaves) {
        int s = src[e];
        int d = dst[e];
        const float4* hrow = (const float4*)(H + (size_t)s * HD);
        float4 v = hrow[lane];                       // 32 lanes x float4 = 128 floats
        float* srow = S + (size_t)d * HD + lane * 4;
        atomicAdd(srow + 0, v.x);
        atomicAdd(srow + 1, v.y);
        atomicAdd(srow + 2, v.z);
        atomicAdd(srow + 3, v.w);
        if (lane == 0) atomicAdd(cnt + d, 1.0f);
    }
}

// ---------------- fused GEMM (fp32 WMMA) + node apply ----------------
// agg = [S ; cnt*H] @ W  (+ cnt*b);  out = sigmoid(agg) + softplus(H)
__global__ __launch_bounds__(256) void mpnn_gemm_apply(const float* __restrict__ S,
                                                       const float* __restrict__ cnt,
                                                       const float* __restrict__ H,
                                                       const float* __restrict__ W,
                                                       const float* __restrict__ bias,
                                                       float* __restrict__ out) {
    __shared__ float Atile[16][LDSS];   // 16 rows x 256 effective-A columns (padded)

    int mtile = blockIdx.x;             // 0..3124, 16 rows each (3125*16 == 50000)
    int t = threadIdx.x;

    // cooperative stage of effective A: [ S_row(0..127) | cnt*H_row(0..127) ]
    for (int i = t; i < 16 * KD; i += 256) {
        int m = i >> 8;                 // / 256
        int k = i & 255;
        int row = mtile * 16 + m;
        float v;
        if (k < HD) v = S[(size_t)row * HD + k];
        else        v = cnt[row] * H[(size_t)row * HD + (k - HD)];
        Atile[m][k] = v;
    }
    __syncthreads();

    int wave  = t >> 5;                 // N-tile 0..7 (8 waves * 16 cols = 128)
    int lane  = t & 31;
    int mrow  = lane & 15;              // A: M = lane%16
    int khalf = (lane >> 4) * 2;        // A/B: K offset 0 (lanes 0-15) or 2 (16-31)
    int ncol  = wave * 16 + (lane & 15);// B/C: N = lane%16 within tile

    v8f acc = {};
    #pragma unroll 8
    for (int k0 = 0; k0 < KD; k0 += 4) {
        int kA = k0 + khalf;
        v2f a, b;
        a.x = Atile[mrow][kA];
        a.y = Atile[mrow][kA + 1];
        b.x = W[(size_t)kA * HD + ncol];          // W row-major [256][128], hot in L2/WGP$
        b.y = W[(size_t)(kA + 1) * HD + ncol];
        // v_wmma_f32_16x16x4_f32: D = A(16x4) * B(4x16) + C
        acc = __builtin_amdgcn_wmma_f32_16x16x4_f32(false, a, false, b,
                                                    (short)0, acc, false, false);
    }

    // epilogue: C layout => VGPR r holds M = r + 8*(lane>=16), N = lane%16
    float bcol  = bias[ncol];
    int   lhalf = lane >> 4;
    #pragma unroll
    for (int r = 0; r < 8; ++r) {
        int row  = mtile * 16 + r + 8 * lhalf;
        float cv = cnt[row];
        float aggv = acc[r] + cv * bcol;          // bias added per edge in reference
        float sg = 1.0f / (1.0f + __expf(-aggv));
        float h  = H[(size_t)row * HD + ncol];
        float sp = fmaxf(h, 0.0f) + log1pf(__expf(-fabsf(h)));  // stable softplus
        out[(size_t)row * HD + ncol] = sg + sp;
    }
}

extern "C" void kernel_launch(void* const* d_in, const int* in_sizes, int n_in,
                              void* d_out, int out_size, void* d_ws, size_t ws_size,
                              hipStream_t stream) {
    const float* H    = (const float*)d_in[0];   // node_feat [50000,128]
    const int*   src  = (const int*)d_in[1];     // [E]
    const int*   dst  = (const int*)d_in[2];     // [E]
    const float* W    = (const float*)d_in[3];   // [256,128]
    const float* bias = (const float*)d_in[4];   // [128]
    float* out = (float*)d_out;
    int E = in_sizes[1];

    float* S   = (float*)d_ws;                   // [50000,128] scatter accumulator
    float* cnt = S + (size_t)NN * HD;            // [50000] in-degree (float)

    int n4 = (NN * HD + NN) / 4;                 // 6,450,000 / 4 exactly
    mpnn_zero<<<1024, 256, 0, stream>>>((float4*)d_ws, n4);
    mpnn_scatter<<<2048, 256, 0, stream>>>(H, src, dst, S, cnt, E);
    mpnn_gemm_apply<<<NN / 16, 256, 0, stream>>>(S, cnt, H, W, bias, out);
}